// GAT_79740362817935
// MI455X (gfx1250) — compile-verified
//
#include <hip/hip_runtime.h>
#include <math.h>

#define NNODES 50000
#define NEDGES 800000
#define GAT_ALPHA 0.2f

typedef __attribute__((ext_vector_type(2))) float v2f;
typedef __attribute__((ext_vector_type(8))) float v8f;

// ---------------------------------------------------------------------------
// Pack W[K x F] -> Bt[Fp x K] (transposed, zero-padded to Fp = colTiles*16).
// Removes all masking + non-contiguous access from the GEMM inner loop.
// ---------------------------------------------------------------------------
__global__ void gat_pack_bt(const float* __restrict__ W,
                            float* __restrict__ Bt, int K, int F, int Fp)
{
    int i = (int)(blockIdx.x * blockDim.x + threadIdx.x);
    if (i >= Fp * K) return;
    int n = i / K;
    int k = i - n * K;
    Bt[i] = (n < F) ? W[(size_t)k * (size_t)F + n] : 0.0f;
}

// ---------------------------------------------------------------------------
// GEMM: C[N x F] = A[N x K] * W[K x F] via V_WMMA_F32_16X16X4_F32, reading W
// through padded/transposed Bt. One wave32 per 16x16 output tile. N = 50000 =
// 3125*16 (no row masking). K is a multiple of 8 for all convs (256/64/40 ->
// K/4 = 64/16/10 even), so the loop runs K/8 iterations with two independent
// accumulator chains (summed at the end) to avoid serializing on WMMA latency.
//
// A 16x4 per-lane layout (ISA 7.12.2): lanes 0-15 hold K=0(v0),K=1(v1);
// lanes 16-31 hold K=2(v0),K=3(v1) -> per-lane contiguous pair at k+2*half.
// B mirrors it; with Bt, lane lr reads column n0+lr contiguously in K.
// C/D: VGPR i = row m0 + i + 8*half, col n0 + lr.
// ---------------------------------------------------------------------------
__global__ void gat_gemm_wmma(const float* __restrict__ A,
                              const float* __restrict__ Bt,
                              float* __restrict__ C,
                              int K, int F, int colTiles, int totalWaves)
{
    int wave = (int)((blockIdx.x * blockDim.x + threadIdx.x) >> 5);
    int lane = (int)(threadIdx.x & 31u);
    if (wave >= totalWaves) return;            // whole-wave uniform exit

    int tm = wave / colTiles;
    int tn = wave - tm * colTiles;
    int m0 = tm * 16;
    int n0 = tn * 16;

    int half = lane >> 4;
    int lr   = lane & 15;
    int nc   = n0 + lr;

    // 8-byte aligned: row bases are K*4B aligned, K % 4 == 0, offsets even.
    const v2f* __restrict__ pa =
        (const v2f*)(A  + (size_t)(m0 + lr) * (size_t)K);
    const v2f* __restrict__ pb =
        (const v2f*)(Bt + (size_t)nc        * (size_t)K);

    v8f acc0 = {};
    v8f acc1 = {};
    int iters = K >> 3;                        // K/8
    for (int it = 0; it < iters; ++it) {
        v2f a0 = pa[4 * it + half];            // floats k .. k+3 (k = 8*it)
        v2f b0 = pb[4 * it + half];
        v2f a1 = pa[4 * it + 2 + half];        // floats k+4 .. k+7
        v2f b1 = pb[4 * it + 2 + half];
        acc0 = __builtin_amdgcn_wmma_f32_16x16x4_f32(
            false, a0, false, b0, (short)0, acc0, false, false);
        acc1 = __builtin_amdgcn_wmma_f32_16x16x4_f32(
            false, a1, false, b1, (short)0, acc1, false, false);
    }
    v8f acc = acc0 + acc1;

    if (nc < F) {                              // column mask only at epilogue
        float* __restrict__ crow =
            C + (size_t)(m0 + 8 * half) * (size_t)F + nc;
#pragma unroll
        for (int i = 0; i < 8; ++i)
            crow[(size_t)i * (size_t)F] = acc[i];
    }
}

// ---------------------------------------------------------------------------
// Per-node attention scores: sS[n] = hW[n,:] . a_s ; sD[n] = hW[n,:] . a_d
// ---------------------------------------------------------------------------
__global__ void gat_scores(const float* __restrict__ hW,
                           const float* __restrict__ a_s,
                           const float* __restrict__ a_d,
                           float* __restrict__ sS, float* __restrict__ sD, int F)
{
    int n = (int)(blockIdx.x * blockDim.x + threadIdx.x);
    if (n >= NNODES) return;
    const float* __restrict__ r = hW + (size_t)n * (size_t)F;
    float s = 0.0f, d = 0.0f;
    for (int f = 0; f < F; ++f) {
        float v = r[f];
        s = fmaf(v, a_s[f], s);
        d = fmaf(v, a_d[f], d);
    }
    sS[n] = s;
    sD[n] = d;
}

// Order-preserving float <-> uint bijection so atomicMax(uint) == float max.
__device__ __forceinline__ unsigned f2mono(float f) {
    unsigned u = __float_as_uint(f);
    return (u & 0x80000000u) ? ~u : (u | 0x80000000u);
}
__device__ __forceinline__ float mono2f(unsigned u) {
    return (u & 0x80000000u) ? __uint_as_float(u & 0x7FFFFFFFu)
                             : __uint_as_float(~u);
}

__global__ void gat_init_md(unsigned* __restrict__ mU, float* __restrict__ denom)
{
    int n = (int)(blockIdx.x * blockDim.x + threadIdx.x);
    if (n >= NNODES) return;
    mU[n]    = 0u;      // below f2mono of any real float: segment-max identity
    denom[n] = 0.0f;
}

__global__ void gat_init_bias(float* __restrict__ out,
                              const float* __restrict__ bias, int F)
{
    int i = (int)(blockIdx.x * blockDim.x + threadIdx.x);
    if (i >= NNODES * F) return;
    out[i] = bias[i % F];
}

// e = leaky_relu(sS[src] + sD[dst]); segment max into mU[dst]
__global__ void gat_edge_score(const float* __restrict__ sS,
                               const float* __restrict__ sD,
                               const int* __restrict__ src,
                               const int* __restrict__ dst,
                               float* __restrict__ ebuf,
                               unsigned* __restrict__ mU)
{
    int e = (int)(blockIdx.x * blockDim.x + threadIdx.x);
    if (e >= NEDGES) return;
    float v = sS[src[e]] + sD[dst[e]];
    v = (v > 0.0f) ? v : GAT_ALPHA * v;
    ebuf[e] = v;
    atomicMax(mU + dst[e], f2mono(v));
}

// ex = exp(e - m[dst]); denom[dst] += ex
__global__ void gat_edge_exp(float* __restrict__ ebuf,
                             const int* __restrict__ dst,
                             const unsigned* __restrict__ mU,
                             float* __restrict__ denom)
{
    int e = (int)(blockIdx.x * blockDim.x + threadIdx.x);
    if (e >= NEDGES) return;
    int d = dst[e];
    float ex = expf(ebuf[e] - mono2f(mU[d]));
    ebuf[e] = ex;
    atomicAdd(denom + d, ex);
}

// out[dst,:] += (ex/denom[dst]) * hW[src,:]   -- one wave32 per edge
__global__ void gat_edge_scatter(const float* __restrict__ ebuf,
                                 const float* __restrict__ denom,
                                 const float* __restrict__ hW,
                                 const int* __restrict__ src,
                                 const int* __restrict__ dst,
                                 float* __restrict__ out, int F)
{
    int gid  = (int)(blockIdx.x * blockDim.x + threadIdx.x);
    int e    = gid >> 5;
    int lane = gid & 31;
    if (e >= NEDGES) return;
    int s = src[e], d = dst[e];
    float att = ebuf[e] / denom[d];
    const float* __restrict__ hs = hW + (size_t)s * (size_t)F;
    float* __restrict__ od = out + (size_t)d * (size_t)F;
    for (int f = lane; f < F; f += 32)
        atomicAdd(od + f, att * hs[f]);
}

__global__ void gat_elu(float* __restrict__ x, int n)
{
    int i = (int)(blockIdx.x * blockDim.x + threadIdx.x);
    if (i >= n) return;
    float v = x[i];
    x[i] = (v > 0.0f) ? v : (expf(v) - 1.0f);
}

__global__ void gat_log_softmax(const float* __restrict__ x,
                                float* __restrict__ out, int F)
{
    int n = (int)(blockIdx.x * blockDim.x + threadIdx.x);
    if (n >= NNODES) return;
    const float* __restrict__ r = x + (size_t)n * (size_t)F;
    float m = -INFINITY;
    for (int f = 0; f < F; ++f) m = fmaxf(m, r[f]);
    float s = 0.0f;
    for (int f = 0; f < F; ++f) s += expf(r[f] - m);
    float ls = logf(s);
    float* __restrict__ o = out + (size_t)n * (size_t)F;
    for (int f = 0; f < F; ++f) o[f] = r[f] - m - ls;
}

// ---------------------------------------------------------------------------
// Host-side orchestration
// ---------------------------------------------------------------------------
static inline int cdiv(int a, int b) { return (a + b - 1) / b; }

static void run_conv(const float* x, int K,
                     const float* W, const float* a_s, const float* a_d,
                     const float* bias, int F,
                     const int* src, const int* dst,
                     float* hW, float* out,
                     float* sS, float* sD, float* denom, unsigned* mU,
                     float* ebuf, float* btbuf, hipStream_t stream)
{
    int colTiles   = cdiv(F, 16);
    int Fp         = colTiles * 16;
    int rowTiles   = NNODES / 16;              // exact: 50000 = 3125*16
    int totalWaves = rowTiles * colTiles;
    int gemmThreads = totalWaves * 32;

    gat_pack_bt<<<cdiv(Fp * K, 256), 256, 0, stream>>>(W, btbuf, K, F, Fp);
    gat_gemm_wmma<<<cdiv(gemmThreads, 256), 256, 0, stream>>>(
        x, btbuf, hW, K, F, colTiles, totalWaves);
    gat_scores<<<cdiv(NNODES, 256), 256, 0, stream>>>(hW, a_s, a_d, sS, sD, F);
    gat_init_md<<<cdiv(NNODES, 256), 256, 0, stream>>>(mU, denom);
    gat_init_bias<<<cdiv(NNODES * F, 256), 256, 0, stream>>>(out, bias, F);
    gat_edge_score<<<cdiv(NEDGES, 256), 256, 0, stream>>>(sS, sD, src, dst, ebuf, mU);
    gat_edge_exp<<<cdiv(NEDGES, 256), 256, 0, stream>>>(ebuf, dst, mU, denom);
    gat_edge_scatter<<<cdiv(NEDGES * 32, 256), 256, 0, stream>>>(
        ebuf, denom, hW, src, dst, out, F);
}

extern "C" void kernel_launch(void* const* d_in, const int* in_sizes, int n_in,
                              void* d_out, int out_size, void* d_ws, size_t ws_size,
                              hipStream_t stream)
{
    (void)in_sizes; (void)n_in; (void)out_size; (void)ws_size;

    const float* feat = (const float*)d_in[0];   // [50000,256]
    const int*   src  = (const int*)d_in[1];     // [800000]
    const int*   dst  = (const int*)d_in[2];     // [800000]
    const float* W1a  = (const float*)d_in[3];   // [256,64]
    const float* W1b  = (const float*)d_in[4];   // [3,64,64]
    const float* a1s  = (const float*)d_in[5];   // [4,64]
    const float* a1d  = (const float*)d_in[6];   // [4,64]
    const float* b1   = (const float*)d_in[7];   // [4,64]
    const float* W2a  = (const float*)d_in[8];   // [64,40]
    const float* W2b  = (const float*)d_in[9];   // [3,40,40]
    const float* a2s  = (const float*)d_in[10];  // [4,40]
    const float* a2d  = (const float*)d_in[11];  // [4,40]
    const float* b2   = (const float*)d_in[12];  // [4,40]

    // Workspace layout (floats): ~42.5 MB total
    const size_t NF = (size_t)NNODES * 64;
    float*    bufA  = (float*)d_ws;
    float*    bufB  = bufA + NF;
    float*    hW    = bufB + NF;
    float*    sS    = hW + NF;
    float*    sD    = sS + NNODES;
    float*    denom = sD + NNODES;
    unsigned* mU    = (unsigned*)(denom + NNODES);
    float*    ebuf  = (float*)(mU + NNODES);     // [800000]
    float*    btbuf = ebuf + NEDGES;             // [<= 64*256]

    float* cur = bufA;
    float* nxt = bufB;

    // -------- Layer 1 (F = 64) --------
    run_conv(feat, 256, W1a, a1s, a1d, b1, 64, src, dst,
             hW, cur, sS, sD, denom, mU, ebuf, btbuf, stream);
    for (int i = 0; i < 3; ++i) {
        run_conv(cur, 64, W1b + (size_t)i * 64 * 64,
                 a1s + (size_t)(i + 1) * 64, a1d + (size_t)(i + 1) * 64,
                 b1 + (size_t)(i + 1) * 64, 64, src, dst,
                 hW, nxt, sS, sD, denom, mU, ebuf, btbuf, stream);
        float* t = cur; cur = nxt; nxt = t;
    }
    gat_elu<<<cdiv(NNODES * 64, 256), 256, 0, stream>>>(cur, NNODES * 64);

    // -------- Layer 2 (F = 40) --------
    run_conv(cur, 64, W2a, a2s, a2d, b2, 40, src, dst,
             hW, nxt, sS, sD, denom, mU, ebuf, btbuf, stream);
    { float* t = cur; cur = nxt; nxt = t; }
    for (int i = 0; i < 3; ++i) {
        run_conv(cur, 40, W2b + (size_t)i * 40 * 40,
                 a2s + (size_t)(i + 1) * 40, a2d + (size_t)(i + 1) * 40,
                 b2 + (size_t)(i + 1) * 40, 40, src, dst,
                 hW, nxt, sS, sD, denom, mU, ebuf, btbuf, stream);
        float* t = cur; cur = nxt; nxt = t;
    }
    gat_elu<<<cdiv(NNODES * 40, 256), 256, 0, stream>>>(cur, NNODES * 40);

    gat_log_softmax<<<cdiv(NNODES, 256), 256, 0, stream>>>(
        cur, (float*)d_out, 40);
}